// CRF_48000554500374
// MI455X (gfx1250) — compile-verified
//
#include <hip/hip_runtime.h>
#include <math.h>

// ---------------------------------------------------------------------------
// CRF Viterbi decode for MI455X (gfx1250, wave32).
//   feats: (B, T, 64) f32   transitions: (64, 64) f32
//   out  : [0..B) path_score f32, [B .. B+B*T) best_path (stored as f32)
//   d_ws : backpointers, B*T*64 bytes (33.5 MB for B=512,T=1024; L2-resident)
// One wave per batch: lane l owns tags {2l, 2l+1}. Max-plus matvec is not
// WMMA-expressible (tropical semiring), so the core is minimal VALU
// (add+cmp+cndmask+max per (i,j)) fed by ds_load_b128 + v_readlane broadcast.
// Transitions are staged to LDS via the CDNA5 Tensor Data Mover.
// ---------------------------------------------------------------------------

#if defined(__has_builtin)
#  if __has_builtin(__builtin_amdgcn_tensor_load_to_lds)
#    define HAVE_TDM 1
#  endif
#endif
#if defined(HAVE_TDM) && __has_include(<hip/amd_detail/amd_gfx1250_TDM.h>)
#  define TDM_6ARG 1   // therock-10.0 headers => 6-arg builtin
#endif

typedef unsigned int u32x4 __attribute__((ext_vector_type(4)));
typedef int          i32x4 __attribute__((ext_vector_type(4)));
typedef int          i32x8 __attribute__((ext_vector_type(8)));

#define NEG_INIT -10000.0f

__device__ __forceinline__ float rl_f32(float v, int lane) {
  return __int_as_float(__builtin_amdgcn_readlane(__float_as_int(v), lane));
}

// --- backtrack chunk helpers (fully unrolled; buffers stay in VGPRs) -------
__device__ __forceinline__ void bt_prefetch(unsigned int (&buf)[32],
                                            const unsigned char* __restrict__ psib,
                                            int p_hi, int lane, int Tm1) {
  const bool top = (p_hi == Tm1);   // first chunk: k==0 has no row (t==T)
#pragma unroll
  for (int k = 0; k < 32; ++k) {
    const int t = p_hi + 1 - k;     // row needed for step p = p_hi - k
    unsigned int v = 0u;
    if (!(k == 0 && top))
      v = *(const unsigned short*)(psib + t * 64 + 2 * lane);
    buf[k] = v;
  }
}

__device__ __forceinline__ void bt_consume(const unsigned int (&buf)[32],
                                           int p_hi, int lane, int Tm1,
                                           int& cur, float* __restrict__ pout) {
  const bool top = (p_hi == Tm1);
  float myval = 0.0f;
#pragma unroll
  for (int k = 0; k < 32; ++k) {
    const int p = p_hi - k;
    if (!(k == 0 && top)) {
      // cur is wave-uniform: readlane gives an SGPR, chain stays short.
      const int word = __builtin_amdgcn_readlane((int)buf[k], cur >> 1);
      cur = (cur & 1) ? ((word >> 8) & 0xFF) : (word & 0xFF);
    }
    if ((p & 31) == lane) myval = (float)cur;   // capture for coalesced store
  }
  pout[(p_hi - 31) + lane] = myval;             // 128B coalesced per chunk
}

__global__ __launch_bounds__(256)
void crf_viterbi_kernel(const float* __restrict__ feats,
                        const float* __restrict__ trans,
                        float* __restrict__ out,
                        unsigned char* __restrict__ psi,
                        int B, int T) {
  // Tq[m*32 + l] = {T[2l][2m], T[2l+1][2m], T[2l][2m+1], T[2l+1][2m+1]}
  __shared__ float4 Tq[32 * 32];     // 16 KB, conflict-free ds_load_b128
  __shared__ float  Traw[64 * 64];   // 16 KB staging

  const int tid  = threadIdx.x;
  const int lane = tid & 31;
  const int wave = tid >> 5;

  // ---- stage transitions into LDS via Tensor Data Mover (CDNA5 TDM) ----
#ifdef HAVE_TDM
  if (tid < 32) {                                    // one wave issues the DMA
    const unsigned int lds_off = (unsigned int)(size_t)(void*)Traw; // addr[31:0] = LDS offset
    const unsigned long long ga = (unsigned long long)(size_t)trans;
    // D# group0: count=1, lds_addr, global_addr[56:0], type=2
    u32x4 g0 = { 1u, lds_off,
                 (unsigned int)(ga & 0xFFFFFFFFull),
                 (unsigned int)(((ga >> 32) & 0x01FFFFFFull) | 0x80000000u) };
    // D# group1: data_size=4B, tensor_dim0=4096, tensor_dim1=1,
    //            tile 4096x1, tensor_dim0_stride=4096
    i32x8 g1 = { (int)0x00020000,   // data_size=2 (4B)
                 (int)0x10000000,   // tensor_dim0 lo16 (0x1000) << 16
                 (int)0x00010000,   // tensor_dim1=1 << 16
                 (int)0x10000000,   // tile_dim0 (0x1000) << 16
                 1,                 // tile_dim1 = 1
                 4096,              // tensor_dim0_stride lo32
                 (int)0x10000000,   // tensor_dim1_stride lo16 << 16
                 0 };
    i32x4 gz = { 0, 0, 0, 0 };
#ifdef TDM_6ARG
    i32x8 gz8 = { 0, 0, 0, 0, 0, 0, 0, 0 };
    __builtin_amdgcn_tensor_load_to_lds(g0, g1, gz, gz, gz8, 0);
#else
    __builtin_amdgcn_tensor_load_to_lds(g0, g1, gz, gz, 0);
#endif
    __builtin_amdgcn_s_wait_tensorcnt(0);
  }
#else
  for (int i = tid; i < 4096; i += blockDim.x) Traw[i] = trans[i];
#endif
  __syncthreads();

  // ---- permute into quad layout for one ds_load_b128 per j-pair ----
  for (int idx = tid; idx < 1024; idx += blockDim.x) {
    const int m = idx >> 5, l = idx & 31;
    float4 q;
    q.x = Traw[(2 * l)     * 64 + 2 * m];
    q.y = Traw[(2 * l + 1) * 64 + 2 * m];
    q.z = Traw[(2 * l)     * 64 + 2 * m + 1];
    q.w = Traw[(2 * l + 1) * 64 + 2 * m + 1];
    Tq[idx] = q;
  }
  __syncthreads();

  const int b = blockIdx.x * (blockDim.x >> 5) + wave;
  if (b >= B) return;

  const float* __restrict__ fb   = feats + (size_t)b * T * 64;
  unsigned char* __restrict__ psib = psi + (size_t)b * T * 64;

  // delta[2*lane], delta[2*lane+1]
  float dA = NEG_INIT, dB = NEG_INIT;

  // ---------------- forward recursion (compute bound, VALU) ----------------
  for (int t = 1; t < T; ++t) {
    // feat for this step: issued early, consumed ~600 cycles later
    const float2 f = *(const float2*)(fb + (size_t)t * 64 + 2 * lane);
    if (t + 8 < T)  // pre-pull the HBM stream into cache (global_prefetch_b8)
      __builtin_prefetch(fb + (size_t)(t + 8) * 64 + 2 * lane, 0, 1);

    float best0 = -__builtin_inff(), best1 = -__builtin_inff();
    int   idx0 = 0, idx1 = 0;
#pragma unroll
    for (int m = 0; m < 32; ++m) {
      const float4 tq = Tq[m * 32 + lane];          // ds_load_b128, no conflicts
      const float djA = rl_f32(dA, m);              // delta[2m]   (SGPR bcast)
      const float djB = rl_f32(dB, m);              // delta[2m+1] (SGPR bcast)
      // candidates in ascending j; strict '>' keeps first argmax (jnp.argmax)
      const float c0 = tq.x + djA;                  // (i=2l,   j=2m)
      const float c1 = tq.y + djA;                  // (i=2l+1, j=2m)
      idx0 = (c0 > best0) ? (2 * m) : idx0;  best0 = fmaxf(best0, c0);
      idx1 = (c1 > best1) ? (2 * m) : idx1;  best1 = fmaxf(best1, c1);
      const float c2 = tq.z + djB;                  // (i=2l,   j=2m+1)
      const float c3 = tq.w + djB;                  // (i=2l+1, j=2m+1)
      idx0 = (c2 > best0) ? (2 * m + 1) : idx0;  best0 = fmaxf(best0, c2);
      idx1 = (c3 > best1) ? (2 * m + 1) : idx1;  best1 = fmaxf(best1, c3);
    }
    dA = best0 + f.x;
    dB = best1 + f.y;
    // packed backpointers: bytes for tags {2l, 2l+1}, coalesced 64B/row
    *(unsigned short*)(psib + (size_t)t * 64 + 2 * lane) =
        (unsigned short)((idx0 & 0xFF) | ((idx1 & 0xFF) << 8));
  }

  // ---------------- final argmax across the wave (ties -> smallest i) ------
  float v = dA; int bi = 2 * lane;
  if (dB > v) { v = dB; bi = 2 * lane + 1; }
#pragma unroll
  for (int off = 16; off; off >>= 1) {
    const float ov = __shfl_xor(v, off, 32);
    const int   oi = __shfl_xor(bi, off, 32);
    if (ov > v || (ov == v && oi < bi)) { v = ov; bi = oi; }
  }
  if (lane == 0) out[b] = v;          // path_score
  int cur = bi;                       // wave-uniform last tag

  float* __restrict__ pout = out + B + (size_t)b * T;

  // ---------------- backtrack --------------------------------------------
  if ((T & 31) == 0) {
    // Double-buffered chunks of 32 psi rows: loads fully pipelined, dependent
    // chain is readlane + byte-extract only (not memory latency).
    unsigned int bufA[32], bufB[32];
    const int Tm1 = T - 1;
    bt_prefetch(bufA, psib, Tm1, lane, Tm1);
    for (int p_hi = Tm1; p_hi >= 31; p_hi -= 64) {
      const bool haveB = (p_hi - 32 >= 31);
      if (haveB) bt_prefetch(bufB, psib, p_hi - 32, lane, Tm1);
      bt_consume(bufA, p_hi, lane, Tm1, cur, pout);
      if (haveB) {
        if (p_hi - 64 >= 31) bt_prefetch(bufA, psib, p_hi - 64, lane, Tm1);
        bt_consume(bufB, p_hi - 32, lane, Tm1, cur, pout);
      }
    }
  } else {
    // generic fallback (uniform serial chase)
    if (lane == 0) pout[T - 1] = (float)cur;
    for (int t = T - 1; t >= 1; --t) {
      cur = psib[(size_t)t * 64 + cur];
      if (lane == 0) pout[t - 1] = (float)cur;
    }
  }
}

extern "C" void kernel_launch(void* const* d_in, const int* in_sizes, int n_in,
                              void* d_out, int out_size, void* d_ws, size_t ws_size,
                              hipStream_t stream) {
  const float* feats = (const float*)d_in[0];   // (B, T, 64) f32
  const float* trans = (const float*)d_in[1];   // (64, 64) f32
  float* out = (float*)d_out;

  const long long bt = (long long)in_sizes[0] / 64;  // B*T
  const int B = out_size - (int)bt;                  // out = B*(1+T)
  const int T = (int)(bt / B);

  unsigned char* psi = (unsigned char*)d_ws;         // needs B*T*64 bytes

  const int waves_per_block = 8;                     // 256 threads
  dim3 block(256);
  dim3 grid((B + waves_per_block - 1) / waves_per_block);
  crf_viterbi_kernel<<<grid, block, 0, stream>>>(feats, trans, out, psi, B, T);
}